// RelationalEncoder_61546881352252
// MI455X (gfx1250) — compile-verified
//
#include <hip/hip_runtime.h>

#define N_NODES   50000
#define N_EDGES   800000
#define HID       96
#define LAT       64
#define NODE_CONT 64
#define EDGE_CAT  16
#define BN_EPS    1e-5f

typedef float v2f __attribute__((ext_vector_type(2)));
typedef float v8f __attribute__((ext_vector_type(8)));

// ---------------------------------------------------------------------------
// Generic f32 WMMA GEMM:  out = opt_relu( A[N,K] @ W^T  (+ A2[N,K2] @ W2^T)  + bias )
// W is [NC, ldw] row-major (PyTorch-style), so B = W^T.
// grid.x = N/16 row tiles; blockDim = (32, NC/16): one wave per 16-col tile.
// All dims multiples of 16/4 -> no guards, EXEC all ones (WMMA requirement).
// ---------------------------------------------------------------------------
__global__ void gemm_wmma_f32(
    const float* __restrict__ A,  int K,  int lda,
    const float* __restrict__ W,  int ldw,
    const float* __restrict__ A2, int K2, int lda2,
    const float* __restrict__ W2, int ldw2,
    const float* __restrict__ bias,
    float* __restrict__ out, int ldo, int relu)
{
    __shared__ float sA [16 * 96];
    __shared__ float sA2[16 * 96];
    const int lane = threadIdx.x;            // 0..31
    const int wv   = threadIdx.y;            // column tile index
    const int nthr = blockDim.x * blockDim.y;
    const int tid  = wv * 32 + lane;
    const int row0 = blockIdx.x * 16;
    const int col0 = wv * 16;
    const int hi   = lane >> 4;              // 0: lanes 0-15, 1: lanes 16-31
    const int lm   = lane & 15;

    // Stage the 16-row A tile(s) in LDS, shared by all column-tile waves.
    for (int i = tid; i < 16 * K; i += nthr)
        sA[i] = A[(long)(row0 + i / K) * lda + (i % K)];
    if (A2)
        for (int i = tid; i < 16 * K2; i += nthr)
            sA2[i] = A2[(long)(row0 + i / K2) * lda2 + (i % K2)];
    __syncthreads();

    // C layout: VGPR r, lanes 0-15 -> row r, lanes 16-31 -> row r+8; N = lane%16.
    v8f c;
    {
        float b = bias ? bias[col0 + lm] : 0.0f;
#pragma unroll
        for (int r = 0; r < 8; ++r) c[r] = b;
    }

    // f32 16x16x4: A 16x4 (2 VGPRs: K = k0+2*hi, k0+2*hi+1), B 4x16 mirrored.
    for (int k0 = 0; k0 < K; k0 += 4) {
        const int ka = k0 + 2 * hi;
        v2f a, b;
        a.x = sA[lm * K + ka];
        a.y = sA[lm * K + ka + 1];
        b.x = W[(col0 + lm) * ldw + ka];      // B[k][n] = W[n][k]
        b.y = W[(col0 + lm) * ldw + ka + 1];
        c = __builtin_amdgcn_wmma_f32_16x16x4_f32(false, a, false, b,
                                                  (short)0, c, false, false);
    }
    if (A2) {
        for (int k0 = 0; k0 < K2; k0 += 4) {
            const int ka = k0 + 2 * hi;
            v2f a, b;
            a.x = sA2[lm * K2 + ka];
            a.y = sA2[lm * K2 + ka + 1];
            b.x = W2[(col0 + lm) * ldw2 + ka];
            b.y = W2[(col0 + lm) * ldw2 + ka + 1];
            c = __builtin_amdgcn_wmma_f32_16x16x4_f32(false, a, false, b,
                                                      (short)0, c, false, false);
        }
    }

#pragma unroll
    for (int r = 0; r < 8; ++r) {
        float v = c[r];
        if (relu) v = fmaxf(v, 0.0f);
        out[(long)(row0 + r + 8 * hi) * ldo + (col0 + lm)] = v;
    }
}

// ---------------------------------------------------------------------------
__global__ void zero_f32(float* __restrict__ p, long n)
{
    long i = (long)blockIdx.x * blockDim.x + threadIdx.x;
    if (i < n) p[i] = 0.0f;
}

// pooled[dst] += edge_embed[cat];  cnt[dst] += 1.   96 threads = 1 channel each.
#define EDGES_PER_BLOCK 128
__global__ void edge_pool(const int* __restrict__ dst, const int* __restrict__ cat,
                          const float* __restrict__ edge_embed,
                          float* __restrict__ pooled, float* __restrict__ cnt, int E)
{
    __shared__ float semb[EDGE_CAT * HID];   // 6 KB embedding table in LDS
    const int tid = threadIdx.x;             // 0..95
    for (int i = tid; i < EDGE_CAT * HID; i += HID) semb[i] = edge_embed[i];
    __syncthreads();

    int e0 = blockIdx.x * EDGES_PER_BLOCK;
    int e1 = e0 + EDGES_PER_BLOCK; if (e1 > E) e1 = E;
    for (int e = e0; e < e1; ++e) {
        const int d = dst[e];                // uniform -> scalar load
        const int ct = cat[e];
        atomicAdd(&pooled[(long)d * HID + tid], semb[ct * HID + tid]);
        if (tid == 0) atomicAdd(&cnt[d], 1.0f);
    }
}

// pooled /= max(cnt,1);  dinv = rsqrt(cnt + 1)   (self-loop degree)
__global__ void finalize_nodes(float* __restrict__ pooled, const float* __restrict__ cnt,
                               float* __restrict__ dinv, int N)
{
    long i = (long)blockIdx.x * blockDim.x + threadIdx.x;
    if (i >= (long)N * HID) return;
    const int n = (int)(i / HID);
    const int c = (int)(i % HID);
    const float ct = cnt[n];
    pooled[i] = pooled[i] / fmaxf(ct, 1.0f);
    if (c == 0) dinv[n] = rsqrtf(ct + 1.0f);
}

// acc = gcn_b + dinv[n]^2 * hw[n]   (GCN bias + self-loop message)
__global__ void gnn_init(const float* __restrict__ hw, const float* __restrict__ dinv,
                         const float* __restrict__ gcn_b_l, float* __restrict__ acc, int N)
{
    long i = (long)blockIdx.x * blockDim.x + threadIdx.x;
    if (i >= (long)N * HID) return;
    const int n = (int)(i / HID);
    const int c = (int)(i % HID);
    const float di = dinv[n];
    acc[i] = gcn_b_l[c] + hw[i] * di * di;
}

// acc[dst] += dinv[src]*dinv[dst] * hw[src]
__global__ void gnn_scatter(const int* __restrict__ src, const int* __restrict__ dst,
                            const float* __restrict__ hw, const float* __restrict__ dinv,
                            float* __restrict__ acc, int E)
{
    const int tid = threadIdx.x;             // 0..95, one channel each
    int e0 = blockIdx.x * EDGES_PER_BLOCK;
    int e1 = e0 + EDGES_PER_BLOCK; if (e1 > E) e1 = E;
    for (int e = e0; e < e1; ++e) {
        const int s = src[e];                // uniform per iteration -> scalar
        const int d = dst[e];
        const float w = dinv[s] * dinv[d];
        atomicAdd(&acc[(long)d * HID + tid], hw[(long)s * HID + tid] * w);
    }
}

// per-channel sum / sumsq partials -> stats[0..95]=sum, stats[96..191]=sumsq
__global__ void bn_stats(const float* __restrict__ h, float* __restrict__ stats, int N)
{
    const int c = threadIdx.x;               // 0..95
    float s = 0.0f, ss = 0.0f;
    for (int n = blockIdx.x; n < N; n += gridDim.x) {
        const float v = h[(long)n * HID + c];
        s += v; ss += v * v;
    }
    atomicAdd(&stats[c], s);
    atomicAdd(&stats[HID + c], ss);
}

// h = relu( (h - mean) * gamma * rsqrt(var+eps) + beta )
__global__ void bn_apply(float* __restrict__ h, const float* __restrict__ stats,
                         const float* __restrict__ gamma, const float* __restrict__ beta, int N)
{
    long i = (long)blockIdx.x * blockDim.x + threadIdx.x;
    if (i >= (long)N * HID) return;
    const int c = (int)(i % HID);
    const float invN = 1.0f / (float)N;
    const float mean = stats[c] * invN;
    const float var  = stats[HID + c] * invN - mean * mean;
    const float sc   = gamma[c] * rsqrtf(var + BN_EPS);
    const float sh   = beta[c] - mean * sc;
    h[i] = fmaxf(h[i] * sc + sh, 0.0f);
}

// ---------------------------------------------------------------------------
extern "C" void kernel_launch(void* const* d_in, const int* in_sizes, int n_in,
                              void* d_out, int out_size, void* d_ws, size_t ws_size,
                              hipStream_t stream)
{
    const float* x_cont     = (const float*)d_in[0];
    const int*   edge_index = (const int*)  d_in[2];
    const int*   edge_cat   = (const int*)  d_in[3];
    const float* w_node     = (const float*)d_in[5];
    const float* b_node     = (const float*)d_in[6];
    const float* edge_embed = (const float*)d_in[7];
    const float* gcn_w      = (const float*)d_in[8];
    const float* gcn_b      = (const float*)d_in[9];
    const float* bn_gamma   = (const float*)d_in[10];
    const float* bn_beta    = (const float*)d_in[11];
    const float* mlp_w1     = (const float*)d_in[12];
    const float* mlp_b1     = (const float*)d_in[13];
    const float* mlp_w2     = (const float*)d_in[14];
    const float* mlp_b2     = (const float*)d_in[15];
    const float* mu_w       = (const float*)d_in[16];
    const float* mu_b       = (const float*)d_in[17];
    const float* lv_w       = (const float*)d_in[18];
    const float* lv_b       = (const float*)d_in[19];

    float* out = (float*)d_out;
    float* ws  = (float*)d_ws;
    const long NH = (long)N_NODES * HID;

    float* h      = ws;                 // [N, 96] current features
    float* hw     = ws + NH;            // [N, 96] h @ W^T / MLP hidden
    float* acc    = ws + 2 * NH;        // [N, 96] GCN aggregation
    float* pooled = ws + 3 * NH;        // [N, 96] pooled edge features
    float* dinv   = ws + 4 * NH;        // [N]
    float* cnt    = dinv + N_NODES;     // [N]
    float* stats  = cnt + N_NODES;      // [192]

    const int* src = edge_index;             // edge_index[0, :]
    const int* dst = edge_index + N_EDGES;   // edge_index[1, :]

    // zero pooled | dinv | cnt | stats (contiguous tail region)
    const long zn = NH + 2L * N_NODES + 2 * HID;
    zero_f32<<<(int)((zn + 255) / 256), 256, 0, stream>>>(pooled, zn);

    // pooled edge features + in-degree counts
    edge_pool<<<(N_EDGES + EDGES_PER_BLOCK - 1) / EDGES_PER_BLOCK, HID, 0, stream>>>(
        dst, edge_cat, edge_embed, pooled, cnt, N_EDGES);
    finalize_nodes<<<(int)((NH + 255) / 256), 256, 0, stream>>>(pooled, cnt, dinv, N_NODES);

    // node embedding: h = x_cont @ w_node^T + b_node
    dim3 gblk(32, HID / 16);   // 6 waves
    gemm_wmma_f32<<<N_NODES / 16, gblk, 0, stream>>>(
        x_cont, NODE_CONT, NODE_CONT, w_node, NODE_CONT,
        nullptr, 0, 0, nullptr, 0, b_node, h, HID, 0);

    for (int l = 0; l < 2; ++l) {
        // hw = h @ gcn_w[l]^T
        gemm_wmma_f32<<<N_NODES / 16, gblk, 0, stream>>>(
            h, HID, HID, gcn_w + (long)l * HID * HID, HID,
            nullptr, 0, 0, nullptr, 0, nullptr, hw, HID, 0);
        // acc = gcn_b + self-loop, then scatter normalized edge messages
        gnn_init<<<(int)((NH + 255) / 256), 256, 0, stream>>>(hw, dinv, gcn_b + l * HID, acc, N_NODES);
        gnn_scatter<<<(N_EDGES + EDGES_PER_BLOCK - 1) / EDGES_PER_BLOCK, HID, 0, stream>>>(
            src, dst, hw, dinv, acc, N_EDGES);
        // hw = relu( [acc, pooled] @ mlp_w1[l]^T + b1 )  (dual K=96 products)
        const float* W1 = mlp_w1 + (long)l * HID * 2 * HID;
        gemm_wmma_f32<<<N_NODES / 16, gblk, 0, stream>>>(
            acc, HID, HID, W1, 2 * HID,
            pooled, HID, HID, W1 + HID, 2 * HID,
            mlp_b1 + l * HID, hw, HID, 1);
        // h = hw @ mlp_w2[l]^T + b2
        gemm_wmma_f32<<<N_NODES / 16, gblk, 0, stream>>>(
            hw, HID, HID, mlp_w2 + (long)l * HID * HID, HID,
            nullptr, 0, 0, nullptr, 0, mlp_b2 + l * HID, h, HID, 0);
        // BatchNorm (batch stats) + ReLU
        zero_f32<<<1, 192, 0, stream>>>(stats, 2 * HID);
        bn_stats<<<256, HID, 0, stream>>>(h, stats, N_NODES);
        bn_apply<<<(int)((NH + 255) / 256), 256, 0, stream>>>(
            h, stats, bn_gamma + l * HID, bn_beta + l * HID, N_NODES);
    }

    // heads: mu, logvar (concatenated flat in d_out)
    dim3 lblk(32, LAT / 16);   // 4 waves
    gemm_wmma_f32<<<N_NODES / 16, lblk, 0, stream>>>(
        h, HID, HID, mu_w, HID, nullptr, 0, 0, nullptr, 0, mu_b, out, LAT, 0);
    gemm_wmma_f32<<<N_NODES / 16, lblk, 0, stream>>>(
        h, HID, HID, lv_w, HID, nullptr, 0, 0, nullptr, 0, lv_b,
        out + (long)N_NODES * LAT, LAT, 0);
}